// HdcLevelEncoder_71691594105142
// MI455X (gfx1250) — compile-verified
//
#include <hip/hip_runtime.h>
#include <hip/hip_bf16.h>
#include <stdint.h>

#define D_DIM  32768
#define TS     64
#define SPLIT  4
#define TCHUNK 16   // 64 timestamps / 4 splits
#define KCHUNK 30   // 120 keys / 4 splits
#define BLK    256

__device__ __constant__ int c_feat_idx[120] = {
    557, 581, 553, 551, 92, 554, 579, 570, 573, 577, 565, 286, 555, 549, 13,
    550, 63, 580, 556, 564, 0, 576, 567, 552, 578, 588, 597, 566, 571, 44,
    572, 574, 14, 582, 381, 594, 4, 593, 218, 25, 84, 592, 3, 591, 547,
    561, 562, 548, 319, 596, 558, 563, 87, 65, 599, 17, 88, 2, 49, 309,
    6, 81, 15, 590, 589, 43, 273, 420, 546, 568, 400, 277, 202, 287, 434,
    435, 423, 431, 301, 417, 412, 205, 179, 327, 176, 442, 172, 450, 391, 163,
    154, 480, 485, 490, 491, 498, 503, 507, 509, 452, 239, 388, 219, 303, 292,
    310, 316, 320, 322, 324, 326, 330, 336, 263, 262, 339, 340, 256, 345, 347};

// 16-byte GCC vector type matching the builtin's pointee type exactly.
typedef int v4i_gcc __attribute__((vector_size(16)));

// ---- CDNA5 async global->LDS copy (ASYNCcnt path) -------------------------
__device__ __forceinline__ void async_copy_b128(const float* gsrc, float* ldst) {
#if defined(__has_builtin) && __has_builtin(__builtin_amdgcn_global_load_async_to_lds_b128)
    typedef __attribute__((address_space(1))) v4i_gcc* gptr_t;   // ptr to AS1 v4i
    typedef __attribute__((address_space(3))) v4i_gcc* lptr_t;   // ptr to AS3 v4i
    __builtin_amdgcn_global_load_async_to_lds_b128(
        (gptr_t)(uintptr_t)gsrc,
        (lptr_t)(uint32_t)(uintptr_t)ldst,   // low 32b of generic == LDS offset
        /*imm offset=*/0, /*cpol=*/0);
#else
    unsigned           loff = (unsigned)(uintptr_t)ldst;
    unsigned long long ga   = (unsigned long long)(uintptr_t)gsrc;
    asm volatile("global_load_async_to_lds_b128 %0, %1, off"
                 :: "v"(loff), "v"(ga) : "memory");
#endif
}

__device__ __forceinline__ void wait_async0() {
#if defined(__has_builtin) && __has_builtin(__builtin_amdgcn_s_wait_asynccnt)
    __builtin_amdgcn_s_wait_asynccnt(0);
#else
    asm volatile("s_wait_asynccnt 0x0" ::: "memory");
#endif
}

__device__ __forceinline__ int level_idx_sig(float v) {
    v = fminf(fmaxf(v, -5.0f), 5.0f);
    int l = (int)rintf((v + 5.0f) / 10.0f * 2047.0f);   // matches (x-lo)/(hi-lo)*(n-1), RNE
    return l < 0 ? 0 : (l > 2047 ? 2047 : l);
}

// ---- main streaming kernel: partial product over t-chunk, partial sum over k-chunk
__global__ __launch_bounds__(BLK) void hdc_partial_kernel(
    const float* __restrict__ input, const float* __restrict__ feat,
    const float* __restrict__ Wx, const float* __restrict__ Wy,
    const float* __restrict__ Wz, const float* __restrict__ Wt,
    const float* __restrict__ keys,
    float* __restrict__ sp, float* __restrict__ fp)
{
    __shared__ alignas(16) float skeys[KCHUNK][BLK];   // 30 KB keys tile
    __shared__ int soffx[TS], soffy[TS], soffz[TS], sofft[TS];
    __shared__ int sfidx[KCHUNK];

    const int tid   = threadIdx.x;
    const int d     = blockIdx.x * BLK + tid;
    const int split = blockIdx.y;
    const int tbase = split * TCHUNK;
    const int kbase = split * KCHUNK;

    // 1) Kick off async staging of this block's keys tile (KCHUNK rows x BLK floats).
    {
        const int sub    = tid >> 6;        // 4 row-groups
        const int lane64 = tid & 63;        // 64 lanes * float4 = 256 floats per row
        for (int kk = sub; kk < KCHUNK; kk += 4) {
            const float* gsrc = keys + (size_t)(kbase + kk) * D_DIM
                                     + (size_t)blockIdx.x * BLK + lane64 * 4;
            async_copy_b128(gsrc, &skeys[kk][lane64 * 4]);
        }
    }

    // 2) Compute gather offsets / thermometer thresholds while copies are in flight.
    if (tid < TS) {
        const int t = tid;
        soffx[t] = level_idx_sig(input[t * 4 + 1]) * D_DIM;
        soffy[t] = level_idx_sig(input[t * 4 + 2]) * D_DIM;
        soffz[t] = level_idx_sig(input[t * 4 + 3]) * D_DIM;
        int it = (int)rintf((float)t / 64.0f * 63.0f);
        it = it < 0 ? 0 : (it > 63 ? 63 : it);
        sofft[t] = it * D_DIM;
    }
    if (tid < KCHUNK) {
        float fv = feat[c_feat_idx[kbase + tid]];
        fv = fminf(fmaxf(fv, -1.0f), 1.0f);
        sfidx[tid] = (int)rintf((fv + 1.0f) * 0.5f * 32768.0f);
    }

    wait_async0();
    __syncthreads();

    // 3) Partial product over this split's 16 timestamps (coalesced gathered rows).
    float s = 1.0f;
#pragma unroll
    for (int tt = 0; tt < TCHUNK; ++tt) {
        const int t = tbase + tt;
        const float xv = Wx[(size_t)soffx[t] + d];
        const float yv = Wy[(size_t)soffy[t] + d];
        const float zv = Wz[(size_t)soffz[t] + d];
        const float wv = Wt[(size_t)sofft[t] + d];
        s *= (xv + yv + zv) * wv;
    }

    // 4) Partial thermometer-masked sum over this split's 30 keys (from LDS).
    float f = 0.0f;
#pragma unroll 6
    for (int kk = 0; kk < KCHUNK; ++kk) {
        const float kv = skeys[kk][tid];
        f += (d < sfidx[kk]) ? kv : -kv;
    }

    sp[(size_t)split * D_DIM + d] = s;
    fp[(size_t)split * D_DIM + d] = f;
}

// ---- combine: final product/sum across splits, binarize -------------------
__global__ __launch_bounds__(BLK) void hdc_combine_kernel(
    const float* __restrict__ sp, const float* __restrict__ fp,
    float* __restrict__ out)
{
    const int d = blockIdx.x * BLK + threadIdx.x;
    float s = sp[d] * sp[D_DIM + d] * sp[2 * D_DIM + d] * sp[3 * D_DIM + d];
    float f = fp[d] + fp[D_DIM + d] + fp[2 * D_DIM + d] + fp[3 * D_DIM + d];
    float c = s + f + s * f;
    out[d] = (c > 0.0f) ? 1.0f : -1.0f;
}

extern "C" void kernel_launch(void* const* d_in, const int* in_sizes, int n_in,
                              void* d_out, int out_size, void* d_ws, size_t ws_size,
                              hipStream_t stream) {
    const float* input = (const float*)d_in[0];   // (64,4)
    const float* feat  = (const float*)d_in[1];   // (600,)
    const float* Wx    = (const float*)d_in[2];   // (2048,32768)
    const float* Wy    = (const float*)d_in[3];
    const float* Wz    = (const float*)d_in[4];
    const float* Wt    = (const float*)d_in[5];   // (64,32768)
    const float* keys  = (const float*)d_in[6];   // (120,32768)
    float*       out   = (float*)d_out;           // (32768,) fp32

    float* sp = (float*)d_ws;                     // SPLIT * D partial products
    float* fp = sp + (size_t)SPLIT * D_DIM;       // SPLIT * D partial sums (total 1 MB)

    dim3 gridB(D_DIM / BLK, SPLIT);               // 128 x 4 = 512 workgroups
    hdc_partial_kernel<<<gridB, BLK, 0, stream>>>(input, feat, Wx, Wy, Wz, Wt, keys, sp, fp);
    hdc_combine_kernel<<<dim3(D_DIM / BLK), BLK, 0, stream>>>(sp, fp, out);
}